// ScorePredictor_13013750907175
// MI455X (gfx1250) — compile-verified
//
#include <hip/hip_runtime.h>
#include <hip/hip_bf16.h>

// Edge dot-product (DGL u_dot_v): score[e] = dot(x[src[e]], x[dst[e]]), D=128, fp32.
// Bandwidth-bound gather; x (51.2MB) is L2-resident on MI455X (192MB L2).
// One wave32 per 16 edges; lane L covers edge (L&15), half-row (L>>4) with b128 loads.
// Cross-lane pair-reduction done exactly in fp32 via V_WMMA_F32_16X16X4_F32 with an
// all-ones A operand: D[m][n] = s_part0[n] + s_part1[n], landing score[n] in lane n.

typedef __attribute__((ext_vector_type(2))) float v2f;
typedef __attribute__((ext_vector_type(8))) float v8f;

#define DFEAT 128

__global__ __launch_bounds__(256) void edge_dot_wmma_kernel(
    const float* __restrict__ x,
    const int*   __restrict__ src,
    const int*   __restrict__ dst,
    float*       __restrict__ out,
    int nFullGroups, int nEdges)
{
    const int lane = threadIdx.x & 31;
    const int wave = threadIdx.x >> 5;
    const int group = blockIdx.x * (blockDim.x >> 5) + wave;

    if (group < nFullGroups) {
        const int eLocal = lane & 15;       // which of the 16 edges
        const int half   = lane >> 4;       // which 64-float half of the row
        const int e      = group * 16 + eLocal;

        // Edge lists are streamed once: non-temporal so they don't evict x from L2.
        const long long si = (long long)__builtin_nontemporal_load(src + e);
        const long long di = (long long)__builtin_nontemporal_load(dst + e);

        const float4* __restrict__ ps =
            (const float4*)(x + si * (long long)DFEAT + half * 64);
        const float4* __restrict__ pd =
            (const float4*)(x + di * (long long)DFEAT + half * 64);

        // 64-element fp32 dot per lane: 16 b128-load pairs, 4 independent FMA chains.
        float acc0 = 0.f, acc1 = 0.f, acc2 = 0.f, acc3 = 0.f;
#pragma unroll
        for (int j = 0; j < 16; j += 4) {
            float4 a0 = ps[j + 0], b0 = pd[j + 0];
            float4 a1 = ps[j + 1], b1 = pd[j + 1];
            float4 a2 = ps[j + 2], b2 = pd[j + 2];
            float4 a3 = ps[j + 3], b3 = pd[j + 3];
            acc0 = fmaf(a0.x, b0.x, fmaf(a0.y, b0.y, fmaf(a0.z, b0.z, fmaf(a0.w, b0.w, acc0))));
            acc1 = fmaf(a1.x, b1.x, fmaf(a1.y, b1.y, fmaf(a1.z, b1.z, fmaf(a1.w, b1.w, acc1))));
            acc2 = fmaf(a2.x, b2.x, fmaf(a2.y, b2.y, fmaf(a2.z, b2.z, fmaf(a2.w, b2.w, acc2))));
            acc3 = fmaf(a3.x, b3.x, fmaf(a3.y, b3.y, fmaf(a3.z, b3.z, fmaf(a3.w, b3.w, acc3))));
        }
        const float s = (acc0 + acc1) + (acc2 + acc3);

        // Exact fp32 reduction via WMMA: A = ones(16x4), B = {s in one K-row pair, 0},
        // D[m][n] = s_part0[n] + s_part1[n] = score[n], broadcast over m.
        v2f a; a[0] = 1.0f; a[1] = 1.0f;
        v2f b; b[0] = s;    b[1] = 0.0f;
        v8f c = {};
        v8f d = __builtin_amdgcn_wmma_f32_16x16x4_f32(
            /*neg_a=*/false, a, /*neg_b=*/false, b,
            /*c_mod=*/(short)0, c, /*reuse_a=*/false, /*reuse_b=*/false);

        // C/D layout: component 0, lane n (0-15) holds score[n]; lanes 16-31 duplicate.
        if (lane < 16) {
            __builtin_nontemporal_store(d[0], out + (long long)group * 16 + lane);
        }
    } else if (group == nFullGroups) {
        // Generic tail (< 16 edges): lane-per-edge plain fp32 dot. (Empty for 1M edges.)
        const int e = nFullGroups * 16 + lane;
        if (e < nEdges) {
            const long long si = (long long)src[e];
            const long long di = (long long)dst[e];
            const float* ps = x + si * (long long)DFEAT;
            const float* pd = x + di * (long long)DFEAT;
            float s = 0.f;
            for (int k = 0; k < DFEAT; ++k) s = fmaf(ps[k], pd[k], s);
            out[e] = s;
        }
    }
}

extern "C" void kernel_launch(void* const* d_in, const int* in_sizes, int n_in,
                              void* d_out, int out_size, void* d_ws, size_t ws_size,
                              hipStream_t stream) {
    const float* x   = (const float*)d_in[0];
    const int*   src = (const int*)d_in[1];
    const int*   dst = (const int*)d_in[2];
    float*       out = (float*)d_out;

    const int nEdges      = in_sizes[1];
    const int nFullGroups = nEdges >> 4;
    const int rem         = nEdges & 15;
    const int totalWaves  = nFullGroups + (rem ? 1 : 0);
    const int wavesPerBlk = 256 / 32;
    const int blocks      = (totalWaves + wavesPerBlk - 1) / wavesPerBlk;
    if (blocks <= 0) return;

    edge_dot_wmma_kernel<<<blocks, 256, 0, stream>>>(x, src, dst, out,
                                                     nFullGroups, nEdges);
}